// Model_50594714747132
// MI455X (gfx1250) — compile-verified
//
#include <hip/hip_runtime.h>
#include <math.h>

// ---------------- problem constants ----------------
#define NFRAMES 128
#define NFILT   512
#define KLEN    512
#define NPADLEN 1536          // zero-padded FFT length
#define NCOEF   769           // rfft bins of length-1536 signal
#define KPAD    784           // padded row stride for spectra (mult of 16)
#define NSAMP   32768

#define TWO_PI_F      6.28318530717958647692f
#define INV_NPAD_F    (1.0f / 1536.0f)
// phase step per bin, in REVOLUTIONS: theta/(2pi) = 256*tanh(p)/769
#define U_SCALE       ((float)(256.0 / 769.0))

typedef float v2f __attribute__((ext_vector_type(2)));
typedef float v8f __attribute__((ext_vector_type(8)));

enum { EPI_SIN = 0, EPI_MUL_LAT = 1, EPI_FINAL = 2 };

// ---------------------------------------------------------------------------
// f32 WMMA tile GEMM over one 16-row M-tile owned by this wave.
// Compile-time KSTEPS/KVALID/NDIM/MODE: branch-free inner loop, full unroll.
//   A: LDS, row-major [128 x lda]
//   W: global, row-major [KVALID x NDIM]; for k >= KVALID the address is
//      clamped (no OOB) and the value selected to 0 (cndmask, no exec branch).
//   V_WMMA_F32_16X16X4_F32 operand layout (ISA 7.12.2):
//     A lane L: m=L&15, holds K = koff, koff+1, koff = (L>>4)*2
//     B lane L: n=L&15, same koff
//     C lane L, vgpr r: row = r + (L>>4)*8, col = L&15
// ---------------------------------------------------------------------------
template <int KSTEPS, int KVALID, int NDIM, int MODE>
__device__ __forceinline__ void gemm_layer(
    const float* __restrict__ Abuf, int lda,
    const float* __restrict__ Wg,  const float* __restrict__ bg,
    float* __restrict__ Obuf,
    const float* __restrict__ latv,
    float* __restrict__ u_out, float* __restrict__ m_out,
    int lane, int mtile)
{
  const int m     = lane & 15;
  const int koff  = (lane >> 4) << 1;
  const int rbase = mtile * 16 + (lane >> 4) * 8;
  const float* Arow = Abuf + (mtile * 16 + m) * lda;

  for (int nt = 0; nt < (NDIM >> 4); ++nt) {
    const int n = (lane & 15) + nt * 16;
    v8f c = {0.f, 0.f, 0.f, 0.f, 0.f, 0.f, 0.f, 0.f};
    #pragma unroll
    for (int ks = 0; ks < KSTEPS; ++ks) {
      const int k0 = ks * 4 + koff;
      v2f a, b;
      a.x = Arow[k0];
      a.y = Arow[k0 + 1];
      if (4 * KSTEPS == KVALID) {              // folds at compile time
        b.x = Wg[k0 * NDIM + n];
        b.y = Wg[(k0 + 1) * NDIM + n];
      } else {                                 // clamped address + select
        const int ka = (k0     < KVALID) ? k0     : 0;
        const int kb = (k0 + 1 < KVALID) ? k0 + 1 : 0;
        const float va = Wg[ka * NDIM + n];
        const float vb = Wg[kb * NDIM + n];
        b.x = (k0     < KVALID) ? va : 0.f;
        b.y = (k0 + 1 < KVALID) ? vb : 0.f;
      }
      c = __builtin_amdgcn_wmma_f32_16x16x4_f32(
              /*neg_a=*/false, a, /*neg_b=*/false, b,
              /*c_mod=*/(short)0, c, /*reuse_a=*/false, /*reuse_b=*/false);
    }
    const float bias = bg[n];
    if (MODE == EPI_SIN) {
      #pragma unroll
      for (int r = 0; r < 8; ++r)
        Obuf[(rbase + r) * NDIM + n] = sinf(c[r] + bias);
    } else if (MODE == EPI_MUL_LAT) {
      const float lv = latv[n];
      #pragma unroll
      for (int r = 0; r < 8; ++r)
        Obuf[(rbase + r) * NDIM + n] = (c[r] + bias) * lv;
    } else { // EPI_FINAL: phase step in revolutions, and relu magnitude
      #pragma unroll
      for (int r = 0; r < 8; ++r) {
        const int   row = rbase + r;
        const float v   = c[r] + bias;
        if (n < NFILT) u_out[row * NFILT + n]           = tanhf(v) * U_SCALE;
        else           m_out[row * NFILT + (n - NFILT)] = fmaxf(v, 0.f);
      }
    }
  }
}

// ---------------------------------------------------------------------------
// Kernel A: whole MLP chain in one workgroup (8 waves, one M-tile each).
// ---------------------------------------------------------------------------
__global__ __launch_bounds__(256) void k_mlp(
    const float* __restrict__ latent,
    const float* __restrict__ pe_W0, const float* __restrict__ pe_b0,
    const float* __restrict__ pe_W1, const float* __restrict__ pe_b1,
    const float* __restrict__ le_W0, const float* __restrict__ le_b0,
    const float* __restrict__ le_W1, const float* __restrict__ le_b1,
    const float* __restrict__ W0, const float* __restrict__ b0,
    const float* __restrict__ W1, const float* __restrict__ b1,
    const float* __restrict__ W2, const float* __restrict__ b2,
    const float* __restrict__ W3, const float* __restrict__ b3,
    const float* __restrict__ W4, const float* __restrict__ b4,
    float* __restrict__ u_out, float* __restrict__ m_out)
{
  __shared__ float P[NFRAMES * 36];       // pos encodings, K padded 33->36
  __shared__ float bufA[NFRAMES * 128];
  __shared__ float bufB[NFRAMES * 128];
  __shared__ float latv[128];
  __shared__ float tmpv[128];

  const int tid  = threadIdx.x;
  const int lane = tid & 31;
  const int wave = tid >> 5;              // 8 waves -> 8 M-tiles of 16 rows

  for (int i = tid; i < NFRAMES * 36; i += 256) P[i] = 0.f;
  __syncthreads();

  // positional encodings: [pos, sin(2^i pos), cos(2^i pos)]_{i=0..15}
  if (tid < NFRAMES) {
    const float pos = -3.14159265358979f + (float)tid * (float)(6.283185307179586 / 127.0);
    P[tid * 36 + 0] = pos;
    #pragma unroll
    for (int i = 0; i < 16; ++i) {
      float s, c;
      sincosf(ldexpf(pos, i), &s, &c);
      P[tid * 36 + 1 + 2 * i] = s;
      P[tid * 36 + 2 + 2 * i] = c;
    }
  }

  // latent embedding MLP (matvec, M=1): sin after layer 0, linear layer 1
  if (tid < 128) {
    float acc = le_b0[tid];
    for (int i = 0; i < 128; ++i) acc = fmaf(latent[i], le_W0[i * 128 + tid], acc);
    tmpv[tid] = sinf(acc);
  }
  __syncthreads();
  if (tid < 128) {
    float acc = le_b1[tid];
    for (int i = 0; i < 128; ++i) acc = fmaf(tmpv[i], le_W1[i * 128 + tid], acc);
    latv[tid] = acc;
  }
  __syncthreads();

  // pos-encoding MLP: sin(P @ peW0 + b), then (· @ peW1 + b) * lat_emb
  gemm_layer<9, 33, 128, EPI_SIN>(P, 36, pe_W0, pe_b0, bufA, nullptr, nullptr, nullptr, lane, wave);
  __syncthreads();
  gemm_layer<32, 128, 128, EPI_MUL_LAT>(bufA, 128, pe_W1, pe_b1, bufB, latv, nullptr, nullptr, lane, wave);
  __syncthreads();
  // main MLP: 4 x sin layers, then linear 128->1024 split into phase/mag
  gemm_layer<32, 128, 128, EPI_SIN>(bufB, 128, W0, b0, bufA, nullptr, nullptr, nullptr, lane, wave);
  __syncthreads();
  gemm_layer<32, 128, 128, EPI_SIN>(bufA, 128, W1, b1, bufB, nullptr, nullptr, nullptr, lane, wave);
  __syncthreads();
  gemm_layer<32, 128, 128, EPI_SIN>(bufB, 128, W2, b2, bufA, nullptr, nullptr, nullptr, lane, wave);
  __syncthreads();
  gemm_layer<32, 128, 128, EPI_SIN>(bufA, 128, W3, b3, bufB, nullptr, nullptr, nullptr, lane, wave);
  __syncthreads();
  gemm_layer<32, 128, 1024, EPI_FINAL>(bufB, 128, W4, b4, nullptr, nullptr, u_out, m_out, lane, wave);
}

// ---------------------------------------------------------------------------
// Kernel B: filter-bank DFT  S[j][k] = sum_tau fb[j][tau] e^{-i 2pi k tau/1536}
// Exact phase via integer (k*tau) mod 1536 + LDS sin/cos tables.
// grid = (4, 512), block = 256; S stored [filter][KPAD] for coalesced k reads.
// ---------------------------------------------------------------------------
__global__ __launch_bounds__(256) void k_spectrum(
    const float* __restrict__ fb, float* __restrict__ Sre, float* __restrict__ Sim)
{
  __shared__ float ctab[NPADLEN];
  __shared__ float stab[NPADLEN];
  __shared__ float frow[KLEN];
  const int tid = threadIdx.x;
  const int j   = blockIdx.y;                       // filter
  const int k   = blockIdx.x * 256 + tid;           // frequency bin

  for (int i = tid; i < NPADLEN; i += 256) {
    float s, c;
    sincosf((float)i * (TWO_PI_F / (float)NPADLEN), &s, &c);
    ctab[i] = c; stab[i] = s;
  }
  for (int i = tid; i < KLEN; i += 256) frow[i] = fb[j * KLEN + i];
  __syncthreads();

  if (k >= NCOEF) return;
  float re = 0.f, im = 0.f;
  int idx = 0;                                      // (k*tau) mod 1536
  for (int tau = 0; tau < KLEN; ++tau) {
    const float v = frow[tau];
    re = fmaf(v, ctab[idx], re);
    im = fmaf(-v, stab[idx], im);
    idx += k; if (idx >= NPADLEN) idx -= NPADLEN;   // k < 1536 -> one subtract
  }
  Sre[j * KPAD + k] = re;
  Sim[j * KPAD + k] = im;
}

// ---------------------------------------------------------------------------
// Kernel C: per-frame modulated spectrum
//   A[f][k] = sum_j m[f][j] * S[j][k] * e^{-i 2pi k u_fj},  u = phase step/2pi
// Phase reduced to [-0.5,0.5] revolutions with one rndne before sincosf,
// so libm's Payne-Hanek huge-argument path is never taken.
// grid = (4, 128), block = 256. m==0 skip is block-uniform (no divergence).
// ---------------------------------------------------------------------------
__global__ __launch_bounds__(256) void k_modspec(
    const float* __restrict__ Sre, const float* __restrict__ Sim,
    const float* __restrict__ urow_g, const float* __restrict__ msc,
    float* __restrict__ Are, float* __restrict__ Aim)
{
  __shared__ float mrow[NFILT];
  __shared__ float urow[NFILT];
  const int tid = threadIdx.x;
  const int f   = blockIdx.y;
  const int k   = blockIdx.x * 256 + tid;

  for (int i = tid; i < NFILT; i += 256) {
    mrow[i] = msc[f * NFILT + i];
    urow[i] = urow_g[f * NFILT + i];
  }
  __syncthreads();

  if (k >= NCOEF) return;
  const float kf = (float)k;
  float are = 0.f, aim = 0.f;
  for (int j = 0; j < NFILT; ++j) {
    const float m = mrow[j];
    if (m == 0.f) continue;                         // uniform across block
    const float sr = Sre[j * KPAD + k];
    const float si = Sim[j * KPAD + k];
    const float p  = kf * urow[j];                  // phase in revolutions
    const float r  = p - rintf(p);                  // [-0.5, 0.5]
    float s, c;
    sincosf(r * TWO_PI_F, &s, &c);                  // |arg| <= pi: fast path
    are = fmaf(m, fmaf(sr, c,  si * s), are);       // Re{S e^{-i k theta}}
    aim = fmaf(m, fmaf(si, c, -sr * s), aim);       // Im{S e^{-i k theta}}
  }
  Are[f * KPAD + k] = are;
  Aim[f * KPAD + k] = aim;
}

// ---------------------------------------------------------------------------
// Kernel D: inverse transform (c2r convention) + Hann window + 1/512 mean.
//   sig[t] = (1/1536)[A0 + 2*sum_{k=1}^{767}(ReA cos - ImA sin)(2pi k t/1536)
//                     + (-1)^t ReA768]
//   wsig[f][t] = (0.5 - 0.5 cos(2pi t/512)) * sig[t] / 512
// grid = 128, block = 256 (each thread handles t and t+256).
// ---------------------------------------------------------------------------
__global__ __launch_bounds__(256) void k_synth(
    const float* __restrict__ Are, const float* __restrict__ Aim,
    float* __restrict__ wsig)
{
  __shared__ float ar[NCOEF];
  __shared__ float ai[NCOEF];
  __shared__ float ctab[NPADLEN];
  __shared__ float stab[NPADLEN];
  const int tid = threadIdx.x;
  const int f   = blockIdx.x;

  for (int i = tid; i < NPADLEN; i += 256) {
    float s, c;
    sincosf((float)i * (TWO_PI_F / (float)NPADLEN), &s, &c);
    ctab[i] = c; stab[i] = s;
  }
  for (int i = tid; i < NCOEF; i += 256) {
    ar[i] = Are[f * KPAD + i];
    ai[i] = Aim[f * KPAD + i];
  }
  __syncthreads();

  #pragma unroll
  for (int half = 0; half < 2; ++half) {
    const int t = tid + half * 256;                 // 0..511
    float acc = ar[0];
    int idx = t;                                    // (k*t) mod 1536 for k=1
    for (int k = 1; k < 768; ++k) {
      acc = fmaf(2.f * ar[k], ctab[idx], acc);
      acc = fmaf(-2.f * ai[k], stab[idx], acc);
      idx += t; if (idx >= NPADLEN) idx -= NPADLEN; // t < 1536 -> one subtract
    }
    acc += (t & 1) ? -ar[768] : ar[768];            // Nyquist, (-1)^t
    const float win = 0.5f - 0.5f * ctab[3 * t];    // hann: 2pi t/512 = 2pi(3t)/1536
    wsig[f * KLEN + t] = acc * INV_NPAD_F * win * (1.0f / (float)NFILT);
  }
}

// ---------------------------------------------------------------------------
// Kernel E: overlap-add, hop 256, truncate to 32768 samples.
// ---------------------------------------------------------------------------
__global__ __launch_bounds__(256) void k_ola(
    const float* __restrict__ wsig, float* __restrict__ out)
{
  const int s  = blockIdx.x * 256 + threadIdx.x;    // 0..32767
  const int f0 = s >> 8;
  const int t0 = s & 255;
  float v = wsig[f0 * KLEN + t0];
  if (f0 > 0) v += wsig[(f0 - 1) * KLEN + t0 + 256];
  out[s] = v;
}

// ---------------------------------------------------------------------------
extern "C" void kernel_launch(void* const* d_in, const int* in_sizes, int n_in,
                              void* d_out, int out_size, void* d_ws, size_t ws_size,
                              hipStream_t stream) {
  (void)in_sizes; (void)n_in; (void)out_size; (void)ws_size;

  const float* latent = (const float*)d_in[0];
  const float* fb     = (const float*)d_in[1];
  const float* pe_W0  = (const float*)d_in[2];
  const float* pe_b0  = (const float*)d_in[3];
  const float* pe_W1  = (const float*)d_in[4];
  const float* pe_b1  = (const float*)d_in[5];
  const float* le_W0  = (const float*)d_in[6];
  const float* le_b0  = (const float*)d_in[7];
  const float* le_W1  = (const float*)d_in[8];
  const float* le_b1  = (const float*)d_in[9];
  const float* W0 = (const float*)d_in[10]; const float* b0 = (const float*)d_in[11];
  const float* W1 = (const float*)d_in[12]; const float* b1 = (const float*)d_in[13];
  const float* W2 = (const float*)d_in[14]; const float* b2 = (const float*)d_in[15];
  const float* W3 = (const float*)d_in[16]; const float* b3 = (const float*)d_in[17];
  const float* W4 = (const float*)d_in[18]; const float* b4 = (const float*)d_in[19];

  float* ws    = (float*)d_ws;
  float* uph   = ws;                               // 128*512 phase steps (rev)
  float* msc   = uph + NFRAMES * NFILT;            // 128*512 relu magnitudes
  float* Sre   = msc + NFRAMES * NFILT;            // 512*KPAD
  float* Sim   = Sre + NFILT * KPAD;               // 512*KPAD
  float* Are   = Sim + NFILT * KPAD;               // 128*KPAD
  float* Aim   = Are + NFRAMES * KPAD;             // 128*KPAD
  float* wsig  = Aim + NFRAMES * KPAD;             // 128*512
  float* out   = (float*)d_out;                    // 32768

  k_mlp<<<1, 256, 0, stream>>>(latent,
      pe_W0, pe_b0, pe_W1, pe_b1, le_W0, le_b0, le_W1, le_b1,
      W0, b0, W1, b1, W2, b2, W3, b3, W4, b4, uph, msc);

  k_spectrum<<<dim3(4, NFILT), 256, 0, stream>>>(fb, Sre, Sim);

  k_modspec<<<dim3(4, NFRAMES), 256, 0, stream>>>(Sre, Sim, uph, msc, Are, Aim);

  k_synth<<<NFRAMES, 256, 0, stream>>>(Are, Aim, wsig);

  k_ola<<<NSAMP / 256, 256, 0, stream>>>(wsig, out);
}